// BLInputLayer_89069031785171
// MI455X (gfx1250) — compile-verified
//
#include <hip/hip_runtime.h>
#include <stdint.h>

// Problem constants (fixed by the reference module config)
#define NB     8                      // batch
#define NL     32768                  // points per sample
#define NC     64                     // feature channels
#define NC4    (NC / 4)               // float4 per row
#define SX     128                    // spatial size per dim
#define TOTAL  (SX * SX * SX)         // 2,097,152 cells
#define ENDV   0xFFFF                 // end-of-chain marker (NL-1 = 32767 < 0xFFFF)

// ---------------------------------------------------------------------------
// CDNA5 async global<->LDS path detection (device pass only; host pass just
// needs to parse). Round-1 diagnostics proved the builtins exist and expect
// (v4i AS1*, v4i AS3*, imm, imm).
// ---------------------------------------------------------------------------
#if defined(__HIP_DEVICE_COMPILE__)
#  if __has_builtin(__builtin_amdgcn_global_load_async_to_lds_b128) && \
      __has_builtin(__builtin_amdgcn_global_store_async_from_lds_b128)
#    define USE_ASYNC_LDS 1
#  else
#    define USE_ASYNC_LDS 0
#    warning "gfx1250 async-to-LDS builtins NOT available on this toolchain; using plain float4 copy"
#  endif
#else
#  define USE_ASYNC_LDS 0
#endif

#if USE_ASYNC_LDS
typedef int v4i __attribute__((ext_vector_type(4)));
typedef __attribute__((address_space(1))) v4i* gv4_ptr;  // global 16B vector ptr
typedef __attribute__((address_space(3))) v4i* lv4_ptr;  // LDS    16B vector ptr

__device__ __forceinline__ void async_wait0() {
#if __has_builtin(__builtin_amdgcn_s_wait_asynccnt)
  __builtin_amdgcn_s_wait_asynccnt(0);
#else
  asm volatile("s_wait_asynccnt 0x0" ::: "memory");
#endif
}
#endif

// ---------------------------------------------------------------------------
// Kernel 1: zero the output and the head table(s). Head value 0 has tag 0,
// which never matches a live tag (tags are b+1 in [1..8]) -> one init total.
// ---------------------------------------------------------------------------
__global__ void init_kernel(float4* __restrict__ out4, long n4,
                            int4* __restrict__ head4, long h4) {
  const long stride = (long)gridDim.x * blockDim.x;
  const long t = (long)blockIdx.x * blockDim.x + threadIdx.x;
  const float4 fz = make_float4(0.f, 0.f, 0.f, 0.f);
  for (long i = t; i < n4; i += stride) out4[i] = fz;
  const int4 iz = make_int4(0, 0, 0, 0);
  for (long i = t; i < h4; i += stride) head4[i] = iz;
}

// ---------------------------------------------------------------------------
// Kernel 2: build per-cell intrusive linked lists with one atomicExch/point.
// head[lin] holds ((b+1)<<16)|l of the last writer; the exchanged-out value's
// tag tells us whether it is a same-batch chain link or a stale entry.
// bFixed >= 0: single shared table (sequential batches). bFixed < 0: per-batch
// tables, batch = blockIdx.y.
// ---------------------------------------------------------------------------
__global__ void pass1_kernel(const int* __restrict__ coords,
                             int* __restrict__ head,
                             int* __restrict__ next,
                             int bFixed) {
  const int b  = (bFixed >= 0) ? bFixed : (int)blockIdx.y;
  int* hd = (bFixed >= 0) ? head : (head + (size_t)b * TOTAL);
  int* nx = (bFixed >= 0) ? next : (next + (size_t)b * NL);

  const int l = blockIdx.x * blockDim.x + threadIdx.x;
  if (l >= NL) return;

  const int* cp = coords + ((size_t)b * NL + l) * 3;
  const int x = cp[0], y = cp[1], z = cp[2];
  const bool valid = (x >= 0) && (y >= 0) && (z >= 0);

  int nv = ENDV;
  if (valid) {
    const long lin = (long)x * (SX * SX) + (long)y * SX + (long)z;
    if (lin >= 0 && lin < TOTAL) {
      const int tagv = ((b + 1) << 16) | l;
      const int old = atomicExch(&hd[lin], tagv);
      nv = ((old >> 16) == (b + 1)) ? (old & ENDV) : ENDV;
    }
    // weird (out-of-grid but valid) lin: solo point, nv stays ENDV
  }
  nx[l] = nv;
}

// ---------------------------------------------------------------------------
// Kernel 3: 16 lanes per point, one float4 slice each. Only the list head
// proceeds. Singleton -> async b128 global->LDS->global row copy (CDNA5 path).
// Multi-member chain -> VGPR float4 accumulation + min-index tracking, one
// plain store. Invalid points: output row stays zero (skipped).
// ---------------------------------------------------------------------------
__global__ void pass2_kernel(const int* __restrict__ coords,
                             const float4* __restrict__ feat4,
                             float4* __restrict__ out4,
                             const int* __restrict__ head,
                             const int* __restrict__ next,
                             int bFixed) {
  const int b = (bFixed >= 0) ? bFixed : (int)blockIdx.y;
  const int* hd = (bFixed >= 0) ? head : (head + (size_t)b * TOTAL);
  const int* nx = (bFixed >= 0) ? next : (next + (size_t)b * NL);

  const int pt   = blockIdx.x * (blockDim.x >> 4) + (threadIdx.x >> 4);
  const int lane = threadIdx.x & 15;

#if USE_ASYNC_LDS
  __shared__ __align__(16) unsigned char lbuf[256 * 16];
#endif

  const int* cp = coords + ((size_t)b * NL + pt) * 3;
  const int x = cp[0], y = cp[1], z = cp[2];
  const bool valid = (x >= 0) && (y >= 0) && (z >= 0);
  if (!valid) return;  // reference contributes zeros for invalid slots

  const long lin = (long)x * (SX * SX) + (long)y * SX + (long)z;
  const bool inr = (lin >= 0 && lin < TOTAL);
  const bool isHead = inr ? (hd[lin] == (((b + 1) << 16) | pt)) : true;
  if (!isHead) return;

  const float4* f4 = feat4 + (size_t)b * NL * NC4;
  float4*       o4 = out4  + (size_t)b * NL * NC4;

  const int n0 = nx[pt];
  if (n0 == ENDV) {
    // ---- singleton fast path (dominant: ~99% of points) ----
#if USE_ASYNC_LDS
    const float4* src = &f4[(size_t)pt * NC4 + lane];
    float4*       dst = &o4[(size_t)pt * NC4 + lane];
    lv4_ptr lp = (lv4_ptr)(&lbuf[(size_t)threadIdx.x * 16]);
    __builtin_amdgcn_global_load_async_to_lds_b128((gv4_ptr)src, lp, 0, 0);
    async_wait0();  // LDS slot filled
    __builtin_amdgcn_global_store_async_from_lds_b128((gv4_ptr)dst, lp, 0, 0);
    // S_ENDPGM performs an implicit wait-idle; no further sync needed.
#else
    o4[(size_t)pt * NC4 + lane] = f4[(size_t)pt * NC4 + lane];
#endif
  } else {
    // ---- collision chain: accumulate in VGPRs, store once at min index ----
    float4 acc = f4[(size_t)pt * NC4 + lane];
    int mn = pt;
    int cur = n0;
    while (cur != ENDV) {
      const float4 v = f4[(size_t)cur * NC4 + lane];
      acc.x += v.x; acc.y += v.y; acc.z += v.z; acc.w += v.w;
      mn = (cur < mn) ? cur : mn;
      cur = nx[cur];
    }
    o4[(size_t)mn * NC4 + lane] = acc;
  }
}

// ---------------------------------------------------------------------------
// Host launcher. Branch on ws_size once (constant across calls -> capture-safe).
//   parallel path : head[NB][TOTAL] + next[NB][NL]  = 65 MB ws, 3 launches
//   sequential    : head[TOTAL] + next[NL]          = 8.13 MB ws, 1 + 2*NB launches
//     (stream ordering serializes pass1(b)/pass2(b) against batch b+1, which
//      the batch-tagged head table requires)
// ---------------------------------------------------------------------------
extern "C" void kernel_launch(void* const* d_in, const int* in_sizes, int n_in,
                              void* d_out, int out_size, void* d_ws, size_t ws_size,
                              hipStream_t stream) {
  (void)in_sizes; (void)n_in; (void)out_size;

  const int*    coords = (const int*)d_in[0];
  const float4* feat4  = (const float4*)d_in[1];
  float4*       out4   = (float4*)d_out;

  const long   nOut4    = (long)NB * NL * NC4;
  const size_t parBytes = (size_t)NB * TOTAL * sizeof(int) + (size_t)NB * NL * sizeof(int);
  const dim3   blk(256);

  if (ws_size >= parBytes) {
    int* head = (int*)d_ws;
    int* next = (int*)((char*)d_ws + (size_t)NB * TOTAL * sizeof(int));
    init_kernel<<<4096, blk, 0, stream>>>(out4, nOut4, (int4*)head, (long)NB * TOTAL / 4);
    pass1_kernel<<<dim3(NL / 256, NB), blk, 0, stream>>>(coords, head, next, -1);
    pass2_kernel<<<dim3(NL / 16,  NB), blk, 0, stream>>>(coords, feat4, out4, head, next, -1);
  } else {
    int* head = (int*)d_ws;
    int* next = head + TOTAL;
    init_kernel<<<4096, blk, 0, stream>>>(out4, nOut4, (int4*)head, (long)TOTAL / 4);
    for (int b = 0; b < NB; ++b) {
      pass1_kernel<<<NL / 256, blk, 0, stream>>>(coords, head, next, b);
      pass2_kernel<<<NL / 16,  blk, 0, stream>>>(coords, feat4, out4, head, next, b);
    }
  }
}